// SimpleGCNNet_8418135900215
// MI455X (gfx1250) — compile-verified
//
#include <hip/hip_runtime.h>
#include <hip/hip_bf16.h>

#define IN_DIM   96
#define N_CLS    10
#define HID_DIM  128
#define CLS_PAD  16

typedef float v2f __attribute__((ext_vector_type(2)));
typedef float v8f __attribute__((ext_vector_type(8)));

// ---------------- utility kernels ----------------

__global__ void zero_f32(float* __restrict__ p, int n) {
    int i = blockIdx.x * blockDim.x + threadIdx.x;
    if (i < n) p[i] = 0.0f;
}

// in-degree via f32 atomics (values are small exact integers in f32)
__global__ void degree_kernel(const int* __restrict__ dst, float* __restrict__ deg, int n_edges) {
    int e = blockIdx.x * blockDim.x + threadIdx.x;
    if (e < n_edges) atomicAdd(&deg[dst[e]], 1.0f);
}

// deg -> norm in place: norm = rsqrt(max(deg,1))
__global__ void norm_kernel(float* __restrict__ d, int n_nodes) {
    int i = blockIdx.x * blockDim.x + threadIdx.x;
    if (i < n_nodes) {
        float v = d[i];
        v = v < 1.0f ? 1.0f : v;
        d[i] = rsqrtf(v);
    }
}

// out[node, :] = in[node, :] * norm[node]^(1 or 2)   (block = 96 threads)
__global__ void scale_rows(const float* __restrict__ in, const float* __restrict__ norm,
                           float* __restrict__ out, int squared) {
    int node = blockIdx.x;
    int f    = threadIdx.x;
    float nv = norm[node];
    if (squared) nv *= nv;
    size_t idx = (size_t)node * IN_DIM + f;
    out[idx] = in[idx] * nv;
}

// y[dst] += g[src] : one wave per edge, lane handles features L, L+32, L+64 (coalesced)
__global__ __launch_bounds__(256) void scatter_kernel(const float* __restrict__ g,
                                                      const int* __restrict__ src,
                                                      const int* __restrict__ dst,
                                                      float* __restrict__ y, int n_edges) {
    int e = blockIdx.x * (blockDim.x >> 5) + (threadIdx.x >> 5);
    if (e >= n_edges) return;
    int lane = threadIdx.x & 31;
    const float* gs = g + (size_t)src[e] * IN_DIM;
    float*       yd = y + (size_t)dst[e] * IN_DIM;
    atomicAdd(&yd[lane +  0], gs[lane +  0]);
    atomicAdd(&yd[lane + 32], gs[lane + 32]);
    atomicAdd(&yd[lane + 64], gs[lane + 64]);
}

// Fold W1@W2 -> Wc[96][16] (padded with zeros), b1@W2 + b2 -> bc[16]
// grid = 97 (96 rows + 1 bias row), block = 16
__global__ void fold_weights(const float* __restrict__ W1, const float* __restrict__ b1,
                             const float* __restrict__ W2, const float* __restrict__ b2,
                             float* __restrict__ Wc, float* __restrict__ bc) {
    int k = blockIdx.x;      // 0..95 weight row, 96 = bias row
    int c = threadIdx.x;     // 0..15
    float s = 0.0f;
    if (c < N_CLS) {
        if (k < IN_DIM) {
            for (int j = 0; j < HID_DIM; ++j)
                s += W1[k * HID_DIM + j] * W2[j * N_CLS + c];
        } else {
            for (int j = 0; j < HID_DIM; ++j)
                s += b1[j] * W2[j * N_CLS + c];
            s += b2[c];
        }
    }
    if (k < IN_DIM) Wc[k * CLS_PAD + c] = s;
    else            bc[c] = s;
}

// out[50000,10] = h4 @ Wc + bc using V_WMMA_F32_16X16X4_F32.
// One wave computes a 16(node) x 16(class-pad) tile; K loop over 96 in steps of 4.
// A 16x4 f32 layout: lanes 0-15 hold row M=lane {K=0 in v0, K=1 in v1};
//                    lanes 16-31 hold row M=lane-16 {K=2 in v0, K=3 in v1}.
// B 4x16 f32 layout: lanes 0-15 hold col N=lane {K=0 in v0, K=1 in v1};
//                    lanes 16-31 hold col N=lane-16 {K=2 in v0, K=3 in v1}.
// C/D: VGPR r -> lanes 0-15: (M=r, N=lane); lanes 16-31: (M=r+8, N=lane-16).
__global__ __launch_bounds__(32) void gemm_wmma_f32(const float* __restrict__ h4,
                                                    const float* __restrict__ Wc,
                                                    const float* __restrict__ bc,
                                                    float* __restrict__ out, int n_nodes) {
    int tile = blockIdx.x;              // 16 rows per tile; n_nodes % 16 == 0
    int lane = threadIdx.x;
    int half = lane >> 4;               // 0 or 1
    int l16  = lane & 15;

    int row = tile * 16 + l16;
    const float* arow = h4 + (size_t)row * IN_DIM;

    v8f c = {0.f, 0.f, 0.f, 0.f, 0.f, 0.f, 0.f, 0.f};

    #pragma unroll
    for (int kk = 0; kk < IN_DIM; kk += 4) {
        int ka = kk + half * 2;
        v2f a, b;
        a.x = arow[ka];
        a.y = arow[ka + 1];
        b.x = Wc[ka * CLS_PAD + l16];
        b.y = Wc[(ka + 1) * CLS_PAD + l16];
        c = __builtin_amdgcn_wmma_f32_16x16x4_f32(false, a, false, b,
                                                  (short)0, c, false, false);
    }

    int   col  = l16;
    float bias = bc[col];
    if (col < N_CLS) {
        #pragma unroll
        for (int r = 0; r < 8; ++r) {
            int orow = tile * 16 + r + half * 8;
            out[(size_t)orow * N_CLS + col] = c[r] + bias;
        }
    }
}

// ---------------- launch ----------------

static inline size_t align_up(size_t x, size_t a) { return (x + a - 1) / a * a; }

extern "C" void kernel_launch(void* const* d_in, const int* in_sizes, int n_in,
                              void* d_out, int out_size, void* d_ws, size_t ws_size,
                              hipStream_t stream) {
    const float* h   = (const float*)d_in[0];
    const int*   src = (const int*)  d_in[1];
    const int*   dst = (const int*)  d_in[2];
    const float* W1  = (const float*)d_in[3];
    const float* b1  = (const float*)d_in[4];
    const float* W2  = (const float*)d_in[5];
    const float* b2  = (const float*)d_in[6];
    float* out = (float*)d_out;

    const int N = in_sizes[0] / IN_DIM;   // 50000
    const int E = in_sizes[1];            // 800000
    const size_t feat = (size_t)N * IN_DIM;

    // workspace carve-out
    char*  ws   = (char*)d_ws;
    size_t off  = 0;
    float* norm = (float*)(ws + off); off = align_up(off + (size_t)N * 4, 1024);
    float* g    = (float*)(ws + off); off = align_up(off + feat * 4, 1024);
    float* y    = (float*)(ws + off); off = align_up(off + feat * 4, 1024);
    float* Wc   = (float*)(ws + off); off = align_up(off + IN_DIM * CLS_PAD * 4, 1024);
    float* bc   = (float*)(ws + off); off = align_up(off + CLS_PAD * 4, 1024);
    (void)ws_size;

    const int TB = 256;

    // degree + norm
    zero_f32<<<(N + TB - 1) / TB, TB, 0, stream>>>(norm, N);
    degree_kernel<<<(E + TB - 1) / TB, TB, 0, stream>>>(dst, norm, E);
    norm_kernel<<<(N + TB - 1) / TB, TB, 0, stream>>>(norm, N);

    // fold dense weights (independent; overlap with graph phase)
    fold_weights<<<IN_DIM + 1, CLS_PAD, 0, stream>>>(W1, b1, W2, b2, Wc, bc);

    // g = h * norm
    scale_rows<<<N, IN_DIM, 0, stream>>>(h, norm, g, 0);

    const int edges_per_block = TB / 32;
    const int sc_grid = (E + edges_per_block - 1) / edges_per_block;
    const int zero_grid = (int)((feat + TB - 1) / TB);

    for (int layer = 0; layer < 4; ++layer) {
        zero_f32<<<zero_grid, TB, 0, stream>>>(y, (int)feat);
        scatter_kernel<<<sc_grid, TB, 0, stream>>>(g, src, dst, y, E);
        // fused post-scale (this layer) + pre-scale (next layer) = norm^2;
        // last layer: post-scale only.
        scale_rows<<<N, IN_DIM, 0, stream>>>(y, norm, g, layer < 3 ? 1 : 0);
    }

    // out = h4 @ Wc + bc   (N % 16 == 0 -> 3125 tiles)
    gemm_wmma_f32<<<N / 16, 32, 0, stream>>>(g, Wc, bc, out, N);
}